// CustomGeneraicGNN_encoder_83726092468975
// MI455X (gfx1250) — compile-verified
//
#include <hip/hip_runtime.h>

// ---------------------------------------------------------------------------
// CDNA5 (gfx1250) GCN encoder: 3x [GEMM(WMMA f32, LDS-staged B) -> edge
// scatter -> BN -> ReLU] -> global mean pool.  wave32 V_WMMA_F32_16X16X4_F32.
// ---------------------------------------------------------------------------

typedef __attribute__((ext_vector_type(2))) float v2f;
typedef __attribute__((ext_vector_type(8))) float v8f;

#define BN_EPS 1e-5f

__device__ __forceinline__ void atomAddF(float* p, float v) {
  // native global_atomic_add_f32 on CDNA
  unsafeAtomicAdd(p, v);
}

// ------------------------------ utility -------------------------------------
__global__ void k_fill(float* __restrict__ p, float v, int n) {
  int i = blockIdx.x * blockDim.x + threadIdx.x;
  if (i < n) p[i] = v;
}

// deg[dst] += 1  (deg pre-filled with 1.0 for self loops)
__global__ void k_deg_add(const int* __restrict__ dst, float* __restrict__ deg, int E) {
  int i = blockIdx.x * blockDim.x + threadIdx.x;
  if (i < E) atomAddF(&deg[dst[i]], 1.0f);
}

// dis = rsqrt(deg) in place; snorm = dis*dis
__global__ void k_rsqrt_snorm(float* __restrict__ dis, float* __restrict__ snorm, int n) {
  int i = blockIdx.x * blockDim.x + threadIdx.x;
  if (i < n) {
    float r = rsqrtf(dis[i]);
    dis[i] = r;
    snorm[i] = r * r;
  }
}

__global__ void k_enorm(const float* __restrict__ dis, const int* __restrict__ src,
                        const int* __restrict__ dst, float* __restrict__ enorm, int E) {
  int i = blockIdx.x * blockDim.x + threadIdx.x;
  if (i < E) enorm[i] = dis[src[i]] * dis[dst[i]];
}

// ------------------------------ WMMA GEMM -----------------------------------
// HW[N,128] = H[N,128] @ W[128,128], fp32, V_WMMA_F32_16X16X4_F32.
// Block = 256 threads = 8 waves.  W staged in LDS in a k-pair interleaved
// layout: element (k,c) -> wlds[(k>>1)*PAIR_STRIDE + 2*c + (k&1)], so each
// lane's B operand (two consecutive k rows, same column) is one contiguous
// 8-byte ds_load_b64 straight into an even-aligned VGPR pair.
// PAIR_STRIDE = 288 floats (== bank +32): the two half-waves of each access
// hit disjoint 32-bank groups -> conflict-free.
#define GEMM_D 128
#define PAIR_STRIDE 288  // 2*D + 32 pad

__global__ __launch_bounds__(256)
void k_gemm_wmma_f32(const float* __restrict__ H, const float* __restrict__ W,
                     float* __restrict__ HW, int N) {
  __shared__ float wlds[(GEMM_D / 2) * PAIR_STRIDE];

  const int tid = threadIdx.x;
  // cooperative staging: 64 k-pairs x 128 cols; two coalesced b32 global
  // loads per element-pair, one ds_store_b64 (stride-2 cols => all 64 banks)
  for (int idx = tid; idx < (GEMM_D / 2) * GEMM_D; idx += 256) {
    int p   = idx >> 7;        // k-pair
    int col = idx & (GEMM_D - 1);
    v2f w;
    w.x = W[(2 * p) * GEMM_D + col];
    w.y = W[(2 * p + 1) * GEMM_D + col];
    *(v2f*)(&wlds[p * PAIR_STRIDE + 2 * col]) = w;
  }
  __syncthreads();

  const int lane  = tid & 31;
  const int wave  = tid >> 5;
  const int khalf = lane >> 4;    // 0: K pair {0,1}, 1: K pair {2,3}
  const int l16   = lane & 15;
  const int m0 = blockIdx.x * 128 + wave * 16;
  if (m0 >= N) return;            // wave-uniform: EXEC stays all-ones

  int rowA = m0 + l16;
  if (rowA >= N) rowA = N - 1;    // clamp (dup read) keeps EXEC full
  const float* __restrict__ Arow = H + (size_t)rowA * GEMM_D;

  v8f c[8];
#pragma unroll
  for (int t = 0; t < 8; ++t) c[t] = (v8f){0.f,0.f,0.f,0.f,0.f,0.f,0.f,0.f};

#pragma unroll 4
  for (int k = 0; k < GEMM_D; k += 4) {
    const int kk = k + 2 * khalf;           // even
    v2f a;
    a.x = Arow[kk];
    a.y = Arow[kk + 1];
    // base of this k-pair's interleaved row, at this lane's column phase
    const float* __restrict__ wp = &wlds[(kk >> 1) * PAIR_STRIDE + 2 * l16];
#pragma unroll
    for (int t = 0; t < 8; ++t) {
      v2f b = *(const v2f*)(wp + t * 32);   // ds_load_b64: (W[kk][col], W[kk+1][col])
      c[t] = __builtin_amdgcn_wmma_f32_16x16x4_f32(
          /*neg_a=*/false, a, /*neg_b=*/false, b,
          /*c_mod=*/(short)0, c[t], /*reuse_a=*/false, /*reuse_b=*/false);
    }
  }

  const int rbase = m0 + 8 * khalf;  // C/D layout: VGPR i -> M=i (+8 hi half)
#pragma unroll
  for (int t = 0; t < 8; ++t) {
    const int col = t * 16 + l16;
#pragma unroll
    for (int i = 0; i < 8; ++i) {
      int r = rbase + i;
      if (r < N) HW[(size_t)r * GEMM_D + col] = c[t][i];
    }
  }
}

// ------------------------- edge gather/scatter-add --------------------------
// one thread per (edge, 4 features): agg[dst] += HW[src] * enorm[e]
__global__ __launch_bounds__(256)
void k_scatter(const float* __restrict__ HW, const float* __restrict__ enorm,
               const int* __restrict__ src, const int* __restrict__ dst,
               float* __restrict__ agg, int E, int D) {
  const int perEdge = D >> 2;  // float4 chunks per edge
  int tid = blockIdx.x * blockDim.x + threadIdx.x;
  if (tid >= E * perEdge) return;
  int e  = tid / perEdge;
  int f4 = (tid - e * perEdge) << 2;
  int s = src[e], d = dst[e];
  float w = enorm[e];
  float4 v = *(const float4*)(HW + (size_t)s * D + f4);
  float* out = agg + (size_t)d * D + f4;
  atomAddF(out + 0, v.x * w);
  atomAddF(out + 1, v.y * w);
  atomAddF(out + 2, v.z * w);
  atomAddF(out + 3, v.w * w);
}

// ------------------- combine + batchnorm column stats -----------------------
// blockDim.x == D; each thread owns one column over a row-chunk.
// h = agg + hw*snorm + bias; accumulate column sum / sumsq.
__global__ void k_combine_stats(const float* __restrict__ HW, const float* __restrict__ AGG,
                                const float* __restrict__ snorm, const float* __restrict__ bias,
                                float* __restrict__ Hout, float* __restrict__ colsum,
                                float* __restrict__ colsq, int N, int D, int rowsPerBlock) {
  const int col = threadIdx.x;
  int r0 = blockIdx.x * rowsPerBlock;
  int r1 = r0 + rowsPerBlock; if (r1 > N) r1 = N;
  const float b = bias[col];
  float s = 0.f, sq = 0.f;
  for (int r = r0; r < r1; ++r) {
    size_t idx = (size_t)r * D + col;
    float v = AGG[idx] + HW[idx] * snorm[r] + b;
    Hout[idx] = v;
    s  += v;
    sq += v * v;
  }
  atomAddF(&colsum[col], s);
  atomAddF(&colsq[col], sq);
}

// one block, D threads: fold stats into per-column scale/shift
__global__ void k_bn_finalize(const float* __restrict__ colsum, const float* __restrict__ colsq,
                              const float* __restrict__ gamma, const float* __restrict__ beta,
                              float* __restrict__ scale, float* __restrict__ shift, int N) {
  int c = threadIdx.x;
  float invN = 1.0f / (float)N;
  float mu  = colsum[c] * invN;
  float var = colsq[c] * invN - mu * mu;
  float sc  = rsqrtf(var + BN_EPS) * gamma[c];
  scale[c] = sc;
  shift[c] = beta[c] - mu * sc;
}

// h = h*scale[col] + shift[col]; optional relu. float4 vectorized.
__global__ __launch_bounds__(256)
void k_bn_apply(float* __restrict__ H, const float* __restrict__ scale,
                const float* __restrict__ shift, int N, int D, int doRelu) {
  int tid = blockIdx.x * blockDim.x + threadIdx.x;
  int total4 = (N * D) >> 2;
  if (tid >= total4) return;
  int f4  = (tid << 2);
  int col = f4 & (D - 1);                    // D is a power of two (128)
  float4 v  = *(float4*)(H + f4);
  float4 sc = *(const float4*)(scale + col);
  float4 sh = *(const float4*)(shift + col);
  v.x = v.x * sc.x + sh.x;
  v.y = v.y * sc.y + sh.y;
  v.z = v.z * sc.z + sh.z;
  v.w = v.w * sc.w + sh.w;
  if (doRelu) {
    v.x = fmaxf(v.x, 0.f); v.y = fmaxf(v.y, 0.f);
    v.z = fmaxf(v.z, 0.f); v.w = fmaxf(v.w, 0.f);
  }
  *(float4*)(H + f4) = v;
}

// ------------------------------- pooling ------------------------------------
__global__ void k_counts(const int* __restrict__ batch, float* __restrict__ counts, int N) {
  int i = blockIdx.x * blockDim.x + threadIdx.x;
  if (i < N) atomAddF(&counts[batch[i]], 1.0f);
}

__global__ __launch_bounds__(256)
void k_pool_sum(const float* __restrict__ H, const int* __restrict__ batch,
                float* __restrict__ out, int N, int D) {
  int tid = blockIdx.x * blockDim.x + threadIdx.x;
  if (tid >= N * D) return;
  int node = tid / D;
  int col  = tid - node * D;
  atomAddF(&out[(size_t)batch[node] * D + col], H[tid]);
}

__global__ void k_pool_div(float* __restrict__ out, const float* __restrict__ counts,
                           int G, int D) {
  int i = blockIdx.x * blockDim.x + threadIdx.x;
  if (i < G * D) out[i] = out[i] / fmaxf(counts[i / D], 1.0f);
}

// ------------------------------- host side ----------------------------------
static inline size_t roundUp32(size_t n) { return (n + 31) & ~(size_t)31; }
static inline int nBlocks(long long total, int bs) { return (int)((total + bs - 1) / bs); }

extern "C" void kernel_launch(void* const* d_in, const int* in_sizes, int n_in,
                              void* d_out, int out_size, void* d_ws, size_t ws_size,
                              hipStream_t stream) {
  const float* x     = (const float*)d_in[0];
  const int*   eidx  = (const int*)d_in[1];
  const int*   batch = (const int*)d_in[2];
  const float* Ws    = (const float*)d_in[3];
  const float* bs    = (const float*)d_in[4];
  const float* gam   = (const float*)d_in[5];
  const float* bet   = (const float*)d_in[6];
  float* out = (float*)d_out;

  const int N = in_sizes[2];
  const int E = in_sizes[1] / 2;
  const int D = in_sizes[0] / N;       // 128
  const int L = in_sizes[3] / (D * D); // 3
  const int G = out_size / D;          // 256

  const int* src = eidx;
  const int* dst = eidx + E;

  // workspace partition (floats, 32-elem aligned regions)
  float* ws = (float*)d_ws;
  float* dis    = ws; ws += roundUp32(N);
  float* snorm  = ws; ws += roundUp32(N);
  float* enorm  = ws; ws += roundUp32(E);
  float* h      = ws; ws += roundUp32((size_t)N * D);
  float* hw     = ws; ws += roundUp32((size_t)N * D);
  float* agg    = ws; ws += roundUp32((size_t)N * D);
  float* colsum = ws; ws += roundUp32(D);
  float* colsq  = ws; ws += roundUp32(D);
  float* scale  = ws; ws += roundUp32(D);
  float* shift  = ws; ws += roundUp32(D);
  float* counts = ws; ws += roundUp32(G);
  (void)ws_size; (void)n_in;

  const int BS = 256;

  // --- degree / norms ---
  k_fill<<<nBlocks(N, BS), BS, 0, stream>>>(dis, 1.0f, N);
  k_deg_add<<<nBlocks(E, BS), BS, 0, stream>>>(dst, dis, E);
  k_rsqrt_snorm<<<nBlocks(N, BS), BS, 0, stream>>>(dis, snorm, N);
  k_enorm<<<nBlocks(E, BS), BS, 0, stream>>>(dis, src, dst, enorm, E);

  const int rowsPerBlock = 256;
  const int statBlocks = nBlocks(N, rowsPerBlock);

  for (int l = 0; l < L; ++l) {
    const float* hin = (l == 0) ? x : h;
    // GEMM: hw = hin @ W[l]   (WMMA f32 16x16x4, W staged in LDS)
    k_gemm_wmma_f32<<<nBlocks(N, 128), 256, 0, stream>>>(hin, Ws + (size_t)l * D * D, hw, N);
    // zero accumulators
    k_fill<<<nBlocks((long long)N * D, BS), BS, 0, stream>>>(agg, 0.0f, N * D);
    k_fill<<<1, D, 0, stream>>>(colsum, 0.0f, D);
    k_fill<<<1, D, 0, stream>>>(colsq, 0.0f, D);
    // edge scatter-add
    k_scatter<<<nBlocks((long long)E * (D >> 2), BS), BS, 0, stream>>>(hw, enorm, src, dst, agg, E, D);
    // combine + bias + BN stats
    k_combine_stats<<<statBlocks, D, 0, stream>>>(hw, agg, snorm, bs + (size_t)l * D,
                                                  h, colsum, colsq, N, D, rowsPerBlock);
    // BN fold + apply (+relu except last)
    k_bn_finalize<<<1, D, 0, stream>>>(colsum, colsq, gam + (size_t)l * D, bet + (size_t)l * D,
                                       scale, shift, N);
    k_bn_apply<<<nBlocks(((long long)N * D) >> 2, BS), BS, 0, stream>>>(h, scale, shift, N, D,
                                                                        (l < L - 1) ? 1 : 0);
  }

  // --- global mean pool ---
  k_fill<<<nBlocks(G * D, BS), BS, 0, stream>>>(out, 0.0f, G * D);
  k_fill<<<nBlocks(G, BS), BS, 0, stream>>>(counts, 0.0f, G);
  k_counts<<<nBlocks(N, BS), BS, 0, stream>>>(batch, counts, N);
  k_pool_sum<<<nBlocks((long long)N * D, BS), BS, 0, stream>>>(h, batch, out, N, D);
  k_pool_div<<<nBlocks(G * D, BS), BS, 0, stream>>>(out, counts, G, D);
}